// DeepTreeLSTM_19172734010037
// MI455X (gfx1250) — compile-verified
//
#include <hip/hip_runtime.h>
#include <hip/hip_bf16.h>
#include <cstdint>

// ---------------------------------------------------------------------------
// DeepTreeLSTM on MI455X (gfx1250, wave32).
// All GEMMs run on V_WMMA_F32_16X16X4_F32 (fp32 A/B, fp32 acc) for fidelity
// with the fp32 reference. One wave = one 16x16 output tile, K-loop of 64
// wmma steps. iou kernels keep 3 accumulators (i/o/u gate column blocks) and
// fuse sigmoid/tanh/cell-update into the epilogue.
// ---------------------------------------------------------------------------

typedef __attribute__((ext_vector_type(2))) float v2f;
typedef __attribute__((ext_vector_type(8))) float v8f;

#define B_SZ   256
#define NNODES 511
#define HS     256

__device__ __forceinline__ v2f ldv2(const float* p) { return *(const v2f*)p; }
__device__ __forceinline__ float sigmoidf(float x) { return 1.0f / (1.0f + expf(-x)); }

// D = A(16x4,f32) * B(4x16,f32) + C(16x16,f32)
// A frag: lane L holds row (L&15), K = 2*(L>>4) + {0,1}  -> one v2f per lane.
// B frag: lane L holds col (L&15), K = 2*(L>>4) + {0,1}  -> one v2f per lane
//   (weights are stored row-major (N,K), so B[k][n] = W[n*K + k] and the lane's
//    two K values are a contiguous float2 in W's row -- same load shape as A).
// C/D: lane L, vgpr v holds row v + 8*(L>>4), col (L&15).
__device__ __forceinline__ v8f wmma4(v2f a, v2f b, v8f c) {
  return __builtin_amdgcn_wmma_f32_16x16x4_f32(false, a, false, b, (short)0, c,
                                               false, false);
}

// ---------------------------------------------------------------------------
// Leaf level: iou = X_leaf * W_iou^T ; gates ; c = i*u + c_in ; h = o*tanh(c)
// rows = B*256 leaves. Writes heap nodes 255..510 of h_all/c_all.
// ---------------------------------------------------------------------------
__global__ void leaf_iou_kernel(const float* __restrict__ X,
                                const float* __restrict__ c_in,
                                const float* __restrict__ W_iou,
                                const float* __restrict__ b_iou,
                                float* __restrict__ h_all,
                                float* __restrict__ c_all) {
  const int r0   = blockIdx.x * 16;
  const int w    = threadIdx.x >> 5;
  const int n0   = (blockIdx.y * 8 + w) * 16;
  const int lane = threadIdx.x & 31;
  const int ml   = lane & 15;
  const int kh   = lane >> 4;

  const int r  = r0 + ml;
  const int bb = r >> 8;       // batch
  const int j  = r & 255;      // leaf index
  const float* ap  = X + ((size_t)bb * NNODES + 255 + j) * HS + 2 * kh;
  const float* bpi = W_iou + (size_t)(n0 + ml) * HS + 2 * kh;
  const float* bpo = W_iou + (size_t)(n0 + 256 + ml) * HS + 2 * kh;
  const float* bpu = W_iou + (size_t)(n0 + 512 + ml) * HS + 2 * kh;

  v8f ai = {}, ao = {}, au = {};
#pragma unroll 8
  for (int k = 0; k < HS; k += 4) {
    v2f a = ldv2(ap + k);
    ai = wmma4(a, ldv2(bpi + k), ai);
    ao = wmma4(a, ldv2(bpo + k), ao);
    au = wmma4(a, ldv2(bpu + k), au);
  }

  const int col = n0 + ml;
  const float bi = b_iou[col], bo = b_iou[256 + col], bu = b_iou[512 + col];
#pragma unroll
  for (int v = 0; v < 8; ++v) {
    const int rg  = r0 + v + 8 * kh;
    const int bb2 = rg >> 8;
    const int j2  = rg & 255;
    const size_t idx = ((size_t)bb2 * NNODES + 255 + j2) * HS + col;
    float iv = sigmoidf(ai[v] + bi);
    float ov = sigmoidf(ao[v] + bo);
    float uv = tanhf(au[v] + bu);
    float cv = iv * uv + c_in[idx];
    h_all[idx] = ov * tanhf(cv);
    c_all[idx] = cv;
  }
}

// ---------------------------------------------------------------------------
// Forget gates + child aggregation for level d:
//   f = sigmoid(ch_h * U_f_w^T + U_f_b); c_agg[parent] = f0*c0 + f1*c1
// rows = B * 2m child rows (m = 2^d parents per batch). Child row pairs
// (2p,2p+1) land in adjacent accumulator VGPRs of the same lane, so the pair
// reduction is in-lane.
// ---------------------------------------------------------------------------
__global__ void level_f_kernel(const float* __restrict__ h_all,
                               const float* __restrict__ c_all,
                               const float* __restrict__ U_f_w,
                               const float* __restrict__ U_f_b,
                               float* __restrict__ cagg,
                               int d) {
  const int twoM = 2 << d;
  const int s1   = twoM - 1;           // first node of child level
  const int r0   = blockIdx.x * 16;
  const int w    = threadIdx.x >> 5;
  const int n0   = (blockIdx.y * 8 + w) * 16;
  const int lane = threadIdx.x & 31;
  const int ml   = lane & 15;
  const int kh   = lane >> 4;

  const int cr = r0 + ml;              // child row id
  const int bb = cr >> (d + 1);
  const int jj = cr & (twoM - 1);
  const float* ap = h_all + ((size_t)bb * NNODES + s1 + jj) * HS + 2 * kh;
  const float* bp = U_f_w + (size_t)(n0 + ml) * HS + 2 * kh;

  v8f acc = {};
#pragma unroll 8
  for (int k = 0; k < HS; k += 4)
    acc = wmma4(ldv2(ap + k), ldv2(bp + k), acc);

  const int col = n0 + ml;
  const float bias = U_f_b[col];
#pragma unroll
  for (int t = 0; t < 4; ++t) {
    const int cr0 = r0 + 2 * t + 8 * kh;
    const int cr1 = cr0 + 1;
    const int b0 = cr0 >> (d + 1), j0 = cr0 & (twoM - 1);
    const int b1 = cr1 >> (d + 1), j1 = cr1 & (twoM - 1);
    float c0 = c_all[((size_t)b0 * NNODES + s1 + j0) * HS + col];
    float c1 = c_all[((size_t)b1 * NNODES + s1 + j1) * HS + col];
    float f0 = sigmoidf(acc[2 * t] + bias);
    float f1 = sigmoidf(acc[2 * t + 1] + bias);
    cagg[(size_t)(cr0 >> 1) * HS + col] = f0 * c0 + f1 * c1;
  }
}

// ---------------------------------------------------------------------------
// iou + cell update for level d:
//   h_tild = h[child0]+h[child1] (built while loading A fragments)
//   iou = h_tild * U_iou^T ; c = sigmoid(i)*tanh(u) + c_agg ; h = sigmoid(o)*tanh(c)
// ---------------------------------------------------------------------------
__global__ void level_iou_kernel(float* __restrict__ h_all,
                                 float* __restrict__ c_all,
                                 const float* __restrict__ U_iou,
                                 const float* __restrict__ b_iou,
                                 const float* __restrict__ cagg,
                                 int d) {
  const int m  = 1 << d;
  const int s0 = m - 1;
  const int s1 = 2 * m - 1;
  const int r0   = blockIdx.x * 16;
  const int w    = threadIdx.x >> 5;
  const int n0   = (blockIdx.y * 8 + w) * 16;
  const int lane = threadIdx.x & 31;
  const int ml   = lane & 15;
  const int kh   = lane >> 4;

  const int r  = r0 + ml;              // parent row id
  const int bb = r >> d;
  const int j  = r & (m - 1);
  const size_t ch0 = ((size_t)bb * NNODES + s1 + 2 * j) * HS;
  const float* ap0 = h_all + ch0 + 2 * kh;
  const float* ap1 = h_all + ch0 + HS + 2 * kh;
  const float* bpi = U_iou + (size_t)(n0 + ml) * HS + 2 * kh;
  const float* bpo = U_iou + (size_t)(n0 + 256 + ml) * HS + 2 * kh;
  const float* bpu = U_iou + (size_t)(n0 + 512 + ml) * HS + 2 * kh;

  v8f ai = {}, ao = {}, au = {};
#pragma unroll 8
  for (int k = 0; k < HS; k += 4) {
    v2f a = ldv2(ap0 + k) + ldv2(ap1 + k);   // h_tild on the fly
    ai = wmma4(a, ldv2(bpi + k), ai);
    ao = wmma4(a, ldv2(bpo + k), ao);
    au = wmma4(a, ldv2(bpu + k), au);
  }

  const int col = n0 + ml;
  const float bi = b_iou[col], bo = b_iou[256 + col], bu = b_iou[512 + col];
#pragma unroll
  for (int v = 0; v < 8; ++v) {
    const int rg  = r0 + v + 8 * kh;
    const int bb2 = rg >> d;
    const int j2  = rg & (m - 1);
    const size_t idx = ((size_t)bb2 * NNODES + s0 + j2) * HS + col;
    float iv = sigmoidf(ai[v] + bi);
    float ov = sigmoidf(ao[v] + bo);
    float uv = tanhf(au[v] + bu);
    float cv = iv * uv + cagg[(size_t)rg * HS + col];
    h_all[idx] = ov * tanhf(cv);
    c_all[idx] = cv;
  }
}

// ---------------------------------------------------------------------------
// x2[b] = [h_root(256) | mean over nodes 1..509 (256) | emo(32)]
// ---------------------------------------------------------------------------
__global__ void reduce_head_kernel(const float* __restrict__ h_all,
                                   const float* __restrict__ emo,
                                   float* __restrict__ x2) {
  const int b = blockIdx.x;
  const int t = threadIdx.x;   // 256 threads == H columns
  const float* base = h_all + (size_t)b * NNODES * HS;
  float s = 0.0f;
  for (int node = 1; node <= 509; ++node) s += base[(size_t)node * HS + t];
  x2[(size_t)b * 544 + t]       = base[t];            // head (root node)
  x2[(size_t)b * 544 + 256 + t] = s * (1.0f / 509.0f);
  if (t < 32) x2[(size_t)b * 544 + 512 + t] = emo[b * 32 + t];
}

// ---------------------------------------------------------------------------
// Tiny head MLP: 544 -> relu 128 -> relu 64 -> sigmoid 4. One block per batch.
// ---------------------------------------------------------------------------
__global__ void head_mlp_kernel(const float* __restrict__ x2,
                                const float* __restrict__ W_in,  const float* __restrict__ b_in,
                                const float* __restrict__ W_mid, const float* __restrict__ b_mid,
                                const float* __restrict__ W_out, const float* __restrict__ b_out,
                                float* __restrict__ out) {
  __shared__ float xs[544];
  __shared__ float y1[128];
  __shared__ float y2[64];
  const int b = blockIdx.x;
  const int t = threadIdx.x;   // 128 threads
  for (int i = t; i < 544; i += 128) xs[i] = x2[(size_t)b * 544 + i];
  __syncthreads();
  {
    float a = b_in[t];
    const float* wrow = W_in + (size_t)t * 544;
    for (int i = 0; i < 544; ++i) a += xs[i] * wrow[i];
    y1[t] = fmaxf(a, 0.0f);
  }
  __syncthreads();
  if (t < 64) {
    float a = b_mid[t];
    const float* wrow = W_mid + (size_t)t * 128;
    for (int i = 0; i < 128; ++i) a += y1[i] * wrow[i];
    y2[t] = fmaxf(a, 0.0f);
  }
  __syncthreads();
  if (t < 4) {
    float a = b_out[t];
    const float* wrow = W_out + (size_t)t * 64;
    for (int i = 0; i < 64; ++i) a += y2[i] * wrow[i];
    out[(size_t)b * 4 + t] = 1.0f / (1.0f + expf(-a));
  }
}

// ---------------------------------------------------------------------------
extern "C" void kernel_launch(void* const* d_in, const int* in_sizes, int n_in,
                              void* d_out, int out_size, void* d_ws, size_t ws_size,
                              hipStream_t stream) {
  (void)in_sizes; (void)n_in; (void)out_size; (void)ws_size;
  const float* X     = (const float*)d_in[0];
  // d_in[1] = h : initial h is never read by the reference math (leaves use X,
  // internal levels use computed children) -> unused.
  const float* c_in  = (const float*)d_in[2];
  const float* emo   = (const float*)d_in[3];
  const float* W_iou = (const float*)d_in[4];
  const float* U_iou = (const float*)d_in[5];
  const float* b_iou = (const float*)d_in[6];
  const float* U_f_w = (const float*)d_in[7];
  const float* U_f_b = (const float*)d_in[8];
  const float* W_in  = (const float*)d_in[9];
  const float* b_in  = (const float*)d_in[10];
  const float* W_mid = (const float*)d_in[11];
  const float* b_mid = (const float*)d_in[12];
  const float* W_out = (const float*)d_in[13];
  const float* b_out = (const float*)d_in[14];
  float* out = (float*)d_out;

  float* ws = (float*)d_ws;
  const size_t HALL = (size_t)B_SZ * NNODES * HS;     // 33,488,896 floats
  float* h_all = ws;
  float* c_all = h_all + HALL;
  float* cagg  = c_all + HALL;                        // max B*128*256 floats
  float* x2    = cagg + (size_t)B_SZ * 128 * HS;      // B*544 floats

  dim3 blk(256);  // 8 waves; wave w covers n-tile blockIdx.y*8+w

  // Leaves: B*256 = 65536 rows -> 4096 row-tiles, 16 n-tiles (2 x 8 waves).
  leaf_iou_kernel<<<dim3(4096, 2), blk, 0, stream>>>(X, c_in, W_iou, b_iou,
                                                     h_all, c_all);
  // Levels d = 7..0 (sequential tree dependency).
  for (int d = 7; d >= 0; --d) {
    int m = 1 << d;
    level_f_kernel  <<<dim3((B_SZ * 2 * m) / 16, 2), blk, 0, stream>>>(
        h_all, c_all, U_f_w, U_f_b, cagg, d);
    level_iou_kernel<<<dim3((B_SZ * m) / 16, 2), blk, 0, stream>>>(
        h_all, c_all, U_iou, b_iou, cagg, d);
  }
  reduce_head_kernel<<<B_SZ, 256, 0, stream>>>(h_all, emo, x2);
  head_mlp_kernel   <<<B_SZ, 128, 0, stream>>>(x2, W_in, b_in, W_mid, b_mid,
                                               W_out, b_out, out);
}